// TensorTrain_79637283603242
// MI455X (gfx1250) — compile-verified
//
#include <hip/hip_runtime.h>

// Tensor-train chain evaluation, D=32 binary cores, rank R=16, B=262144.
// One wave32 handles TWO tiles of 16 batch rows each; every chain step does
// four V_WMMA_F32_16X16X4_F32 chains (2 tiles x {core^0, core^1}) that share
// the B operands (and their register repacking), with the per-row binary
// select folded into the LDS transpose reload that converts each D-layout
// result back into A-layout for the next step.

#define TT_D 32
#define TT_R 16
#define WAVES_PER_BLOCK 8
#define TILES_PER_WAVE 2
#define ROWS_PER_WAVE (16 * TILES_PER_WAVE)    // 32 batch rows per wave
#define LDS_ROW_STRIDE 18                      // 16 cols + pad; even -> b64-aligned col pairs
#define LDS_BUF_FLOATS (16 * LDS_ROW_STRIDE)   // 288 floats per candidate buffer
#define LDS_PER_TILE (2 * LDS_BUF_FLOATS)      // two candidate buffers per tile

typedef float v2f __attribute__((ext_vector_type(2)));
typedef float v8f __attribute__((ext_vector_type(8)));
typedef int   v4i __attribute__((ext_vector_type(4)));

__global__ __launch_bounds__(256) void tt_chain_wmma_f32(
    const int*   __restrict__ X,           // [B][32]
    const float* __restrict__ core_first,  // [1][R][2] -> flat [R][2]
    const float* __restrict__ cores_mid,   // [30][R][R][2]
    const float* __restrict__ core_last,   // [R][1][2] -> flat [R][2]
    float*       __restrict__ out)         // [B]
{
    __shared__ float lds[WAVES_PER_BLOCK][TILES_PER_WAVE * LDS_PER_TILE];

    const int tid  = threadIdx.x;
    const int wave = tid >> 5;
    const int lane = tid & 31;
    const int row  = lane & 15;   // batch row within tile (M index)
    const int half = lane >> 4;   // lane half selects K sub-pair / M+8 rows

    const int base = (blockIdx.x * WAVES_PER_BLOCK + wave) * ROWS_PER_WAVE;

    float* wbuf = lds[wave];

    // ---- pack each tile-row's 32 binary digits into one mask (streamed NT) ----
    unsigned xm[TILES_PER_WAVE];
    #pragma unroll
    for (int t = 0; t < TILES_PER_WAVE; ++t) {
        const v4i* xr = (const v4i*)(X + (size_t)(base + 16 * t + row) * TT_D);
        unsigned m = 0u;
        #pragma unroll
        for (int q = 0; q < 8; ++q) {
            v4i v = __builtin_nontemporal_load(xr + q);
            m |= (unsigned)(v.x & 1) << (4 * q + 0);
            m |= (unsigned)(v.y & 1) << (4 * q + 1);
            m |= (unsigned)(v.z & 1) << (4 * q + 2);
            m |= (unsigned)(v.w & 1) << (4 * q + 3);
        }
        xm[t] = m;
    }

    // ---- initial carries in WMMA A-layout ----
    // chunk c (K-slab 4c..4c+3): lane holds K = 4c + 2*half + j, row M = lane%16
    float a[TILES_PER_WAVE][4][2];
    #pragma unroll
    for (int t = 0; t < TILES_PER_WAVE; ++t) {
        const unsigned x0 = xm[t] & 1u;
        #pragma unroll
        for (int c = 0; c < 4; ++c)
            #pragma unroll
            for (int j = 0; j < 2; ++j) {
                int k = 4 * c + 2 * half + j;
                a[t][c][j] = core_first[k * 2 + x0];
            }
    }

    // ---- chain through the 30 middle cores ----
    #pragma unroll 1
    for (int ci = 0; ci < TT_D - 2; ++ci) {
        // B operands, shared by both tiles: lane holds column N = row;
        // VGPR j holds K = 4c + 2*half + j. (e=0,e=1) loaded together.
        const float* Bc = cores_mid + ci * (TT_R * TT_R * 2);
        float2 bv[4][2];
        #pragma unroll
        for (int c = 0; c < 4; ++c)
            #pragma unroll
            for (int j = 0; j < 2; ++j) {
                int k = 4 * c + 2 * half + j;
                bv[c][j] = *(const float2*)(Bc + k * 32 + row * 2);
            }
        v2f b0[4], b1[4];
        #pragma unroll
        for (int c = 0; c < 4; ++c) {
            b0[c] = (v2f){ bv[c][0].x, bv[c][1].x };
            b1[c] = (v2f){ bv[c][0].y, bv[c][1].y };
        }

        // Four candidate products, K=16 chained as 4x V_WMMA_F32_16X16X4_F32.
        v8f p0[TILES_PER_WAVE], p1[TILES_PER_WAVE];
        #pragma unroll
        for (int t = 0; t < TILES_PER_WAVE; ++t) { p0[t] = (v8f){}; p1[t] = (v8f){}; }

        #pragma unroll
        for (int c = 0; c < 4; ++c) {
            #pragma unroll
            for (int t = 0; t < TILES_PER_WAVE; ++t) {
                v2f av = { a[t][c][0], a[t][c][1] };
                p0[t] = __builtin_amdgcn_wmma_f32_16x16x4_f32(
                            false, av, false, b0[c], (short)0, p0[t], false, false);
                p1[t] = __builtin_amdgcn_wmma_f32_16x16x4_f32(
                            false, av, false, b1[c], (short)0, p1[t], false, false);
            }
        }

        // D-layout -> LDS (both candidate results, per tile).
        #pragma unroll
        for (int t = 0; t < TILES_PER_WAVE; ++t) {
            float* tb = wbuf + t * LDS_PER_TILE;
            #pragma unroll
            for (int v = 0; v < 8; ++v) {
                int r = v + 8 * half;
                tb[r * LDS_ROW_STRIDE + row]                  = p0[t][v];
                tb[LDS_BUF_FLOATS + r * LDS_ROW_STRIDE + row] = p1[t][v];
            }
        }

        // Per-row branch select during the A-layout reload. Same-wave LDS ops
        // are in-order; region is wave-private, so no barriers needed.
        #pragma unroll
        for (int t = 0; t < TILES_PER_WAVE; ++t) {
            const unsigned sel = (xm[t] >> (ci + 1)) & 1u;
            const float* src = wbuf + t * LDS_PER_TILE + sel * LDS_BUF_FLOATS
                               + row * LDS_ROW_STRIDE + 2 * half;
            #pragma unroll
            for (int c = 0; c < 4; ++c) {
                float2 tv = *(const float2*)(src + 4 * c);
                a[t][c][0] = tv.x;
                a[t][c][1] = tv.y;
            }
        }
    }

    // ---- last core: per-lane partial dot in A-layout, combine lane halves ----
    #pragma unroll
    for (int t = 0; t < TILES_PER_WAVE; ++t) {
        const unsigned xl = (xm[t] >> 31) & 1u;
        float acc = 0.f;
        #pragma unroll
        for (int c = 0; c < 4; ++c)
            #pragma unroll
            for (int j = 0; j < 2; ++j) {
                int k = 4 * c + 2 * half + j;
                acc += a[t][c][j] * core_last[k * 2 + xl];
            }
        acc += __shfl_xor(acc, 16, 32);   // add the other half's K-partial
        if (half == 0)
            __builtin_nontemporal_store(acc, out + base + 16 * t + row);
    }
}

extern "C" void kernel_launch(void* const* d_in, const int* in_sizes, int n_in,
                              void* d_out, int out_size, void* d_ws, size_t ws_size,
                              hipStream_t stream) {
    (void)in_sizes; (void)n_in; (void)d_ws; (void)ws_size;
    const int*   X  = (const int*)d_in[0];
    const float* cf = (const float*)d_in[1];
    const float* cm = (const float*)d_in[2];
    const float* cl = (const float*)d_in[3];
    float* out = (float*)d_out;

    const int rows_per_block = ROWS_PER_WAVE * WAVES_PER_BLOCK; // 256
    const int blocks = out_size / rows_per_block;               // 262144/256 = 1024
    tt_chain_wmma_f32<<<blocks, 256, 0, stream>>>(X, cf, cm, cl, out);
}